// NeighborhoodAttention_3710851743835
// MI455X (gfx1250) — compile-verified
//
#include <hip/hip_runtime.h>
#include <math.h>

// ---------------------------------------------------------------------------
// Neighborhood attention (B=2, 56x56, C=128, 4 heads, 7x7 window), fp32.
// GEMMs (qkv proj 617 MF + out proj 206 MF) run on V_WMMA_F32_16X16X4_F32
// (fp32 reference -> fp32 matrix pipe, no precision loss). Each wave now
// carries TWO independent accumulator chains (16x32 macro-tile) so the
// multicycle XDL WMMA pipe overlaps two D->C chains instead of stalling on
// one. Per-pixel gather attention uses coalesced loads + wave32 reductions.
// ---------------------------------------------------------------------------

typedef __attribute__((ext_vector_type(2))) float v2f;
typedef __attribute__((ext_vector_type(8))) float v8f;

#define BB     2
#define HH     56
#define WWID   56
#define CCH    128
#define NHEAD  4
#define DHEAD  32
#define KNB    7
#define KK     49          // 7*7 neighbors
#define POSW   13          // 2*K-1
#define NPIX   (BB * HH * WWID)

__device__ __forceinline__ float waveSum(float v) {
#pragma unroll
  for (int m = 16; m >= 1; m >>= 1) v += __shfl_xor(v, m, 32);
  return v;
}
__device__ __forceinline__ float waveMax(float v) {
#pragma unroll
  for (int m = 16; m >= 1; m >>= 1) v = fmaxf(v, __shfl_xor(v, m, 32));
  return v;
}

// ---------------------------------------------------------------------------
// C[M x N] = A[M x K] * B[K x N] + bias[N], fp32 WMMA 16x16x4.
// 8 waves / block; each wave owns a 16x32 macro-tile = two 16x16 WMMA tiles
// sharing the A fragment (two independent accumulator chains for XDL ILP).
// A-frag (16x4 f32): lane%16 = M row, lane/16 selects K pair {0,1}|{2,3}.
// B-frag (4x16 f32): lane%16 = N col, same K-pair split.
// C/D (16x16 f32, 8 VGPRs): N = lane%16, M = vgpr + 8*(lane/16).
// Requires: M % 16 == 0, N % 32 == 0, K % 4 == 0.
// ---------------------------------------------------------------------------
__global__ __launch_bounds__(256) void na_gemm_bias_wmma(
    const float* __restrict__ A, const float* __restrict__ B,
    const float* __restrict__ bias, float* __restrict__ C,
    int M, int N, int K) {
  const int wave = threadIdx.x >> 5;
  const int lane = threadIdx.x & 31;
  const int tilesN = N >> 5;               // macro-tiles of width 32
  const int tilesM = M >> 4;
  const int tile = blockIdx.x * 8 + wave;
  if (tile >= tilesM * tilesN) return;     // wave-uniform: EXEC stays all-1s

  const int tm = (tile / tilesN) << 4;
  const int tn = (tile % tilesN) << 5;
  const int ln = lane & 15;
  const int lh = lane >> 4;                // 0 or 1 -> K sub-pair

  const float bv0 = bias[tn + ln];         // same column for all 8 acc rows
  const float bv1 = bias[tn + 16 + ln];
  v8f acc0, acc1;
#pragma unroll
  for (int i = 0; i < 8; ++i) { acc0[i] = bv0; acc1[i] = bv1; }

  const float* __restrict__ Arow  = A + (size_t)(tm + ln) * K + (lh << 1);
  const float* __restrict__ Bcol0 = B + (size_t)(lh << 1) * N + (tn + ln);
  const float* __restrict__ Bcol1 = Bcol0 + 16;

#pragma unroll 4
  for (int k0 = 0; k0 < K; k0 += 4) {
    v2f a = *(const v2f*)(Arow + k0);      // 8B-aligned: k0+2*lh is even
    v2f b0, b1;
    b0.x = Bcol0[(size_t)k0 * N];
    b0.y = Bcol0[(size_t)k0 * N + N];
    b1.x = Bcol1[(size_t)k0 * N];
    b1.y = Bcol1[(size_t)k0 * N + N];
    acc0 = __builtin_amdgcn_wmma_f32_16x16x4_f32(
        false, a, false, b0, (short)0, acc0, false, false);
    acc1 = __builtin_amdgcn_wmma_f32_16x16x4_f32(
        false, a, false, b1, (short)0, acc1, false, false);
  }

#pragma unroll
  for (int i = 0; i < 8; ++i) {
    const size_t row = (size_t)(tm + i + (lh << 3)) * N;
    C[row + tn + ln]      = acc0[i];
    C[row + tn + 16 + ln] = acc1[i];
  }
}

// ---------------------------------------------------------------------------
// Per-pixel neighborhood attention.
// block = 128 threads = 4 waves (one per head); lane = channel d in [0,32).
// qkv layout: [NPIX][384] with q|k|v each [head][32].
// Bias index derived from reference _bias_coords after the [::-1] flip:
//   idx = (3-(y-clamp(y,3,52)) + i)*13 + (3-(x-clamp(x,3,52)) + j)
// Neighbor gather uses edge clamping (jnp.pad mode='edge').
// ---------------------------------------------------------------------------
__global__ __launch_bounds__(128) void na_attn_kernel(
    const float* __restrict__ qkv, const float* __restrict__ pos_bias,
    float* __restrict__ out) {
  const int n    = blockIdx.x;             // pixel id over B*H*W
  const int head = threadIdx.x >> 5;
  const int lane = threadIdx.x & 31;
  const int p    = n % (HH * WWID);
  const int y    = p / WWID;
  const int x    = p % WWID;
  const int base = n - p;                  // batch offset in pixels
  const float scale = 0.17677669529663687f; // 1/sqrt(32)

  const float ql =
      qkv[(size_t)n * (3 * CCH) + head * DHEAD + lane] * scale;

  const int fy = 3 - y + min(max(y, 3), HH - 4);   // bias row base
  const int fx = 3 - x + min(max(x, 3), WWID - 4); // bias col base
  const float* __restrict__ pb = pos_bias + head * (POSW * POSW);

  // ---- scores: lane t owns score[t] (s0) and score[t+32] (s1) ----
  float s0 = -1e30f, s1 = -1e30f;
#pragma unroll 1
  for (int t = 0; t < KK; ++t) {
    const int i  = t / KNB, j = t % KNB;
    const int yy = min(max(y + i - 3, 0), HH - 1);
    const int xx = min(max(x + j - 3, 0), WWID - 1);
    const int pos = base + yy * WWID + xx;
    const float kd =
        qkv[(size_t)pos * (3 * CCH) + CCH + head * DHEAD + lane];
    float sc = waveSum(ql * kd);           // broadcast to all lanes
    sc += pb[(fy + i) * POSW + (fx + j)];
    if (t < 32) { if (lane == t)      s0 = sc; }
    else        { if (lane == t - 32) s1 = sc; }
  }

  // ---- softmax over 49 (lane-resident) ----
  const float m  = waveMax(fmaxf(s0, s1));
  const float e0 = __expf(s0 - m);
  const float e1 = (lane < KK - 32) ? __expf(s1 - m) : 0.0f;
  const float inv = 1.0f / waveSum(e0 + e1);
  const float a0 = e0 * inv;
  const float a1 = e1 * inv;

  // ---- out[d] = sum_t attn[t] * v[pos_t][d] ----
  float acc = 0.0f;
#pragma unroll 1
  for (int t = 0; t < KK; ++t) {
    const int i  = t / KNB, j = t % KNB;
    const int yy = min(max(y + i - 3, 0), HH - 1);
    const int xx = min(max(x + j - 3, 0), WWID - 1);
    const int pos = base + yy * WWID + xx;
    const float w = (t < 32) ? __shfl(a0, t, 32) : __shfl(a1, t - 32, 32);
    const float vd =
        qkv[(size_t)pos * (3 * CCH) + 2 * CCH + head * DHEAD + lane];
    acc = fmaf(w, vd, acc);
  }
  out[(size_t)n * CCH + head * DHEAD + lane] = acc;
}

// ---------------------------------------------------------------------------
// inputs: x, qkv_w, qkv_b, pos_bias, out_w, out_b (all fp32)
// ws: qkv buffer (NPIX*384 f32) + attn-out buffer (NPIX*128 f32) ~ 12.9 MB
// ---------------------------------------------------------------------------
extern "C" void kernel_launch(void* const* d_in, const int* in_sizes, int n_in,
                              void* d_out, int out_size, void* d_ws, size_t ws_size,
                              hipStream_t stream) {
  const float* x        = (const float*)d_in[0];
  const float* qkv_w    = (const float*)d_in[1];
  const float* qkv_b    = (const float*)d_in[2];
  const float* pos_bias = (const float*)d_in[3];
  const float* out_w    = (const float*)d_in[4];
  const float* out_b    = (const float*)d_in[5];
  float* out = (float*)d_out;

  float* qkv  = (float*)d_ws;                      // NPIX * 384
  float* attn = qkv + (size_t)NPIX * (3 * CCH);    // NPIX * 128

  const int M = NPIX;                              // 6272 = 392 * 16

  // 1) qkv = x @ qkv_w + qkv_b   (6272 x 384, K=128)
  {
    const int tiles = (M >> 4) * ((3 * CCH) >> 5); // 392 * 12
    na_gemm_bias_wmma<<<(tiles + 7) / 8, 256, 0, stream>>>(
        x, qkv_w, qkv_b, qkv, M, 3 * CCH, CCH);
  }

  // 2) neighborhood attention -> attn (6272 x 128)
  na_attn_kernel<<<M, 128, 0, stream>>>(qkv, pos_bias, attn);

  // 3) out = attn @ out_w + out_b  (6272 x 128, K=128)
  {
    const int tiles = (M >> 4) * (CCH >> 5);       // 392 * 4
    na_gemm_bias_wmma<<<(tiles + 7) / 8, 256, 0, stream>>>(
        attn, out_w, out_b, out, M, CCH, CCH);
  }
}